// TaskIncrementalPrompt_20091857011402
// MI455X (gfx1250) — compile-verified
//
#include <hip/hip_runtime.h>
#include <hip/hip_bf16.h>
#include <math.h>

// Problem constants
#define B_    256
#define S_    197
#define D_    768
#define POOL_ 100
#define LEN_  5
#define PPT_  10
#define TOPK_ 5
#define ROWS_ (TOPK_*LEN_ + S_)   // 222

typedef __attribute__((ext_vector_type(2))) float v2f;
typedef __attribute__((ext_vector_type(8))) float v8f;

// ---------------------------------------------------------------------------
// L2 normalize rows of [rows, 768]: one block per row, 256 threads, 3 elems/t
// ---------------------------------------------------------------------------
__global__ void l2norm_rows_kernel(const float* __restrict__ src,
                                   float* __restrict__ dst) {
  const int row = blockIdx.x, t = threadIdx.x;
  __shared__ float red[256];
  const float* s = src + (size_t)row * D_;
  float x0 = s[t], x1 = s[t + 256], x2 = s[t + 512];
  red[t] = x0 * x0 + x1 * x1 + x2 * x2;
  __syncthreads();
  for (int k = 128; k > 0; k >>= 1) {
    if (t < k) red[t] += red[t + k];
    __syncthreads();
  }
  float sq = red[0];
  sq = (sq < 1e-12f) ? 1e-12f : sq;
  float r = 1.0f / sqrtf(sq);
  float* d = dst + (size_t)row * D_;
  d[t] = x0 * r; d[t + 256] = x1 * r; d[t + 512] = x2 * r;
}

// ---------------------------------------------------------------------------
// similarity[256,10] = Xnorm[256,768] . Knorm[10,768]^T via V_WMMA_F32_16X16X4_F32
// One wave32 per 16-row M tile (16 blocks x 32 threads). N padded to 16 by
// clamping key rows (garbage columns 10..15 are never stored; WMMA columns
// are independent so valid columns are unaffected).
// A layout (f32 16x4): lanes 0-15 -> M=lane, v0=K0,v1=K1 ; lanes 16-31 -> K2,K3
// B layout (f32 4x16): lanes 0-15 -> N=lane, v0=K0,v1=K1 ; lanes 16-31 -> K2,K3
// C/D layout: VGPR j, lanes 0-15 -> M=j, lanes 16-31 -> M=j+8, N=lane%16
// ---------------------------------------------------------------------------
__global__ void sim_wmma_kernel(const float* __restrict__ xnorm,
                                const float* __restrict__ knorm,
                                float* __restrict__ sim) {
  const int mtile = blockIdx.x;          // 0..15
  const int lane  = threadIdx.x;         // 0..31
  const int half  = lane >> 4;           // 0 or 1 (selects K pair)
  const int l     = lane & 15;
  const int arow  = mtile * 16 + l;                // batch row
  const int brow  = (l < PPT_) ? l : (PPT_ - 1);   // clamp key row (pad)
  const float* aptr = xnorm + (size_t)arow * D_ + 2 * half;
  const float* bptr = knorm + (size_t)brow * D_ + 2 * half;
  v8f c = {};
  #pragma unroll 8
  for (int d0 = 0; d0 < D_; d0 += 4) {
    v2f a = *(const v2f*)(aptr + d0);
    v2f b = *(const v2f*)(bptr + d0);
    c = __builtin_amdgcn_wmma_f32_16x16x4_f32(false, a, false, b,
                                              (short)0, c, false, false);
  }
  if (l < PPT_) {
    #pragma unroll
    for (int j = 0; j < 8; ++j) {
      int m = mtile * 16 + j + 8 * half;
      sim[(size_t)m * PPT_ + l] = c[j];
    }
  }
}

// ---------------------------------------------------------------------------
// Per-batch: stable top-5 of 10, write idx (as float) + idx ints to ws,
// gather selected_key rows, and deterministic partial dot for reduce_sim.
// ---------------------------------------------------------------------------
__global__ void topk_select_kernel(const float* __restrict__ sim,
                                   const float* __restrict__ knorm,
                                   const float* __restrict__ xnorm,
                                   float* __restrict__ idx_out,
                                   int*   __restrict__ idx_ws,
                                   float* __restrict__ selkey,
                                   float* __restrict__ partial) {
  const int b = blockIdx.x, t = threadIdx.x;
  __shared__ int   sel[TOPK_];
  __shared__ float red[256];
  if (t == 0) {
    float v[PPT_];
    bool used[PPT_];
    for (int i = 0; i < PPT_; ++i) { v[i] = sim[b * PPT_ + i]; used[i] = false; }
    for (int k = 0; k < TOPK_; ++k) {
      int best = 0; float bv = -INFINITY; bool found = false;
      for (int i = 0; i < PPT_; ++i) {
        if (!used[i] && (!found || v[i] > bv)) { bv = v[i]; best = i; found = true; }
      }
      used[best] = true;
      sel[k] = best;
      idx_out[b * TOPK_ + k] = (float)best;
      idx_ws[b * TOPK_ + k]  = best;
    }
  }
  __syncthreads();
  float acc = 0.f;
  for (int e = t; e < TOPK_ * D_; e += 256) {
    int k = e / D_, d = e - k * D_;
    float kv = knorm[(size_t)sel[k] * D_ + d];
    selkey[(size_t)b * (TOPK_ * D_) + e] = kv;
    acc += kv * xnorm[(size_t)b * D_ + d];
  }
  red[t] = acc;
  __syncthreads();
  for (int k = 128; k > 0; k >>= 1) {
    if (t < k) red[t] += red[t + k];
    __syncthreads();
  }
  if (t == 0) partial[b] = red[0];
}

// ---------------------------------------------------------------------------
// reduce_sim = sum(partial[0..255]) / B  (fixed-order -> deterministic)
// ---------------------------------------------------------------------------
__global__ void reduce_sim_kernel(const float* __restrict__ partial,
                                  float* __restrict__ out) {
  __shared__ float red[256];
  const int t = threadIdx.x;
  red[t] = partial[t];
  __syncthreads();
  for (int k = 128; k > 0; k >>= 1) {
    if (t < k) red[t] += red[t + k];
    __syncthreads();
  }
  if (t == 0) out[0] = red[0] / (float)B_;
}

// ---------------------------------------------------------------------------
// prompted_embedding[b, r, :] :
//   r <  25 : prompt[idx[b, r/5], r%5, :]
//   r >= 25 : x_embed[b, r-25, :]
// One block per (b,r) row; 256 threads x 3 dwords. Dest base is only 4B
// aligned (odd float offset) so dword granularity; consecutive lanes on
// consecutive dwords = fully coalesced 128B per instruction.
// ---------------------------------------------------------------------------
__global__ void concat_kernel(const float* __restrict__ prompt,
                              const float* __restrict__ x_embed,
                              const int*   __restrict__ idx_ws,
                              float* __restrict__ out) {
  const int blk = blockIdx.x;            // b*222 + r
  const int b = blk / ROWS_;
  const int r = blk - b * ROWS_;
  const float* src;
  if (r < TOPK_ * LEN_) {
    const int j = r / LEN_;
    const int l = r - j * LEN_;
    const int p = idx_ws[b * TOPK_ + j];
    src = prompt + ((size_t)p * LEN_ + l) * D_;
  } else {
    src = x_embed + ((size_t)b * S_ + (r - TOPK_ * LEN_)) * D_;
  }
  float* dst = out + ((size_t)b * ROWS_ + r) * D_;
  const int t = threadIdx.x;
  dst[t]       = src[t];
  dst[t + 256] = src[t + 256];
  dst[t + 512] = src[t + 512];
}

// ---------------------------------------------------------------------------
// Launch
// ---------------------------------------------------------------------------
extern "C" void kernel_launch(void* const* d_in, const int* in_sizes, int n_in,
                              void* d_out, int out_size, void* d_ws, size_t ws_size,
                              hipStream_t stream) {
  const float* x_embed = (const float*)d_in[0];   // [256,197,768]
  const float* cls     = (const float*)d_in[1];   // [256,768]
  const float* prompt  = (const float*)d_in[2];   // [100,5,768]
  const float* pkey    = (const float*)d_in[3];   // [10,768]

  float* out   = (float*)d_out;
  float* o_idx = out;                 // 1280
  float* o_pn  = out + 1280;          // 7680
  float* o_xn  = out + 8960;          // 196608
  float* o_sim = out + 205568;        // 2560
  float* o_sk  = out + 208128;        // 983040
  float* o_rs  = out + 1191168;       // 1
  float* o_pe  = out + 1191169;       // 43646976

  float* ws_partial = (float*)d_ws;                      // 256 floats
  int*   ws_idx     = (int*)((char*)d_ws + 1024);        // 1280 ints

  l2norm_rows_kernel<<<PPT_, 256, 0, stream>>>(pkey, o_pn);
  l2norm_rows_kernel<<<B_,   256, 0, stream>>>(cls,  o_xn);
  sim_wmma_kernel<<<16, 32, 0, stream>>>(o_xn, o_pn, o_sim);
  topk_select_kernel<<<B_, 256, 0, stream>>>(o_sim, o_pn, o_xn,
                                             o_idx, ws_idx, o_sk, ws_partial);
  reduce_sim_kernel<<<1, 256, 0, stream>>>(ws_partial, o_rs);
  concat_kernel<<<B_ * ROWS_, 256, 0, stream>>>(prompt, x_embed, ws_idx, o_pe);
}